// CustomReshapeLayer_24154896072774
// MI455X (gfx1250) — compile-verified
//
#include <hip/hip_runtime.h>
#include <stdint.h>

// out[b, r, c] = (c >= r) ? x[b, base(r) + c - r] : 0
// B=64, M=2048, N_TRIU = M*(M+1)/2 = 2098176
#define MATSIZE 2048u
#define NTRIU   2098176ull

#ifndef __has_builtin
#define __has_builtin(x) 0
#endif

#if __has_builtin(__builtin_amdgcn_global_store_async_from_lds_b128)
#define USE_ASYNC_ZERO 1
#else
#define USE_ASYNC_ZERO 0
#endif

typedef float f32x4 __attribute__((ext_vector_type(4)));
// Builtin's parameter type per hipcc diagnostic: int __attribute__((vector_size(16))) AS*
typedef int v4i __attribute__((vector_size(16)));
#define AS_GLOBAL __attribute__((address_space(1)))
#define AS_LDS    __attribute__((address_space(3)))

// One block per (batch, row). r/base/src-base are wave-uniform (SALU),
// so per-lane work is just column addressing + the memory ops.
__global__ __launch_bounds__(256) void triu_row_kernel(const float* __restrict__ x,
                                                       float* __restrict__ out) {
#if USE_ASYNC_ZERO
    // 32B of zeros in LDS (each store reads 16B); streamed to the below-diagonal region via the async
    // LDS->global engine (ASYNCcnt path, VGPR-free stores).
    __shared__ __align__(16) float zbuf[8];
    if (threadIdx.x < 8) zbuf[threadIdx.x] = 0.0f;
    __syncthreads();
#endif

    uint32_t blk = blockIdx.x;          // [0, 64*2048)
    uint32_t r   = blk & 2047u;         // row (scalar)
    uint32_t b   = blk >> 11;           // batch (scalar)

    // base(r) = r*M - r*(r-1)/2 = r*(2M - r + 1)/2  (max ~2.1e6, fits u32; scalar)
    uint32_t base = (r * (2u * MATSIZE - r + 1u)) >> 1;
    const float* __restrict__ src = x + (uint64_t)b * NTRIU + (uint64_t)(base - r);
    float* __restrict__ orow = out + ((uint64_t)blk << 11);   // row base, 8KB aligned

    uint32_t t4 = threadIdx.x << 2;     // 0..1020, step 4

#pragma unroll
    for (uint32_t k = 0; k < 2; ++k) {
        uint32_t c = t4 + (k << 10);            // columns [c, c+4)
        float* outp = orow + c;                 // 16B aligned

        if (c + 3u < r) {
            // Strictly below the diagonal: stream zeros.
#if USE_ASYNC_ZERO
            __builtin_amdgcn_global_store_async_from_lds_b128(
                (AS_GLOBAL v4i*)outp,
                (AS_LDS v4i*)(&zbuf[0]),
                0, 0);
#else
            f32x4 z = (f32x4)0.0f;
            __builtin_nontemporal_store(z, (f32x4*)outp);
#endif
        } else {
            f32x4 v;
            if (c >= r) {
                // Fully inside the triangle: 4 consecutive source dwords.
                // (dword-aligned + coalesced; row shift forbids b128 loads)
                const float* p = src + c;
                v.x = __builtin_nontemporal_load(p + 0);
                v.y = __builtin_nontemporal_load(p + 1);
                v.z = __builtin_nontemporal_load(p + 2);
                v.w = __builtin_nontemporal_load(p + 3);
            } else {
                // Diagonal-straddling vec4 (at most one lane per row).
                v.x = (c + 0u >= r) ? src[c + 0u] : 0.0f;
                v.y = (c + 1u >= r) ? src[c + 1u] : 0.0f;
                v.z = (c + 2u >= r) ? src[c + 2u] : 0.0f;
                v.w = (c + 3u >= r) ? src[c + 3u] : 0.0f;
            }
            __builtin_nontemporal_store(v, (f32x4*)outp);
        }
    }
    // S_ENDPGM performs an implicit wait-idle, draining ASYNCcnt/STOREcnt.
}

extern "C" void kernel_launch(void* const* d_in, const int* in_sizes, int n_in,
                              void* d_out, int out_size, void* d_ws, size_t ws_size,
                              hipStream_t stream) {
    const float* x = (const float*)d_in[0];
    float* out = (float*)d_out;

    // One block per (batch, row): 64 * 2048 = 131072 blocks, 256 threads,
    // each thread writes two float4s (columns 4t and 4t+1024).
    dim3 grid(64u * 2048u);
    dim3 block(256u);
    hipLaunchKernelGGL(triu_row_kernel, grid, block, 0, stream, x, out);
}